// PointPillarsScatter_11888469475596
// MI455X (gfx1250) — compile-verified
//
#include <hip/hip_runtime.h>
#include <cstdint>
#include <cstddef>

// Problem constants (match reference)
#define KNY 496
#define KNX 432
#define KC  64
#define KB  8
#define KN  131072
#define KSPATIAL (KNY * KNX)                   // 214272
#define KSLOTS   (KB * KSPATIAL)               // 1714176 (divisible by 4)
#define KOUTCHUNKS (KB * KC * KSPATIAL / 4)    // 27,426,816 x 16B chunks
#define KOUTGROUPS (KOUTCHUNKS / 4)            // 6,856,704 x 64B groups

typedef __attribute__((ext_vector_type(4))) float v4f;
typedef __attribute__((ext_vector_type(4))) int   v4i;   // pointee type for async builtin

// ---------------------------------------------------------------------------
// Kernel 1: zero the 438.8 MB canvas via the CDNA5 async DMA store path.
// Each block zeroes a 4KB LDS buffer once, then streams b128 async stores
// (LDS -> global, ASYNCcnt-tracked). Each lane owns 64B contiguous per trip
// (wave covers 2KB); the +16/+32/+48 deltas go through the instruction's
// immediate offset field so the GVS scale_offset addressing form is kept.
// ---------------------------------------------------------------------------
__global__ __launch_bounds__(256) void zero_fill_async(float* __restrict__ out,
                                                       unsigned ngroups) {
    __shared__ v4i zbuf[256];                 // 4 KB of zeros
    unsigned t = threadIdx.x;
    v4i z = {};
    zbuf[t] = z;
    __syncthreads();

    v4i* lsrc  = &zbuf[t];                    // each lane sources its own 16B
    v4i* gbase = (v4i*)out;                   // uniform SGPR base
    unsigned stride = gridDim.x * blockDim.x;
    for (unsigned g = blockIdx.x * blockDim.x + t; g < ngroups; g += stride) {
        v4i* gdst = gbase + (size_t)g * 4;    // byte offset g*64 = (g*4)*16
        __builtin_amdgcn_global_store_async_from_lds_b128(gdst, lsrc,  0, 0);
        __builtin_amdgcn_global_store_async_from_lds_b128(gdst, lsrc, 16, 0);
        __builtin_amdgcn_global_store_async_from_lds_b128(gdst, lsrc, 32, 0);
        __builtin_amdgcn_global_store_async_from_lds_b128(gdst, lsrc, 48, 0);
    }
    __builtin_amdgcn_s_wait_asynccnt(0);      // drain before wave exit
}

// ---------------------------------------------------------------------------
// Kernel 2: winner[slot] = -1 (b128 vectorized)
// ---------------------------------------------------------------------------
__global__ __launch_bounds__(256) void init_winner(v4i* __restrict__ winner) {
    unsigned i = blockIdx.x * 256u + threadIdx.x;
    if (i < (KSLOTS / 4)) {
        v4i m = {-1, -1, -1, -1};
        winner[i] = m;
    }
}

// ---------------------------------------------------------------------------
// Kernel 3: per voxel, compute flat slot, record it, and claim the slot with
// atomicMax so duplicate indices resolve deterministically (max voxel wins).
// ---------------------------------------------------------------------------
__global__ __launch_bounds__(256) void claim_kernel(const int4* __restrict__ coors,
                                                    int* __restrict__ winner,
                                                    int* __restrict__ slot_arr) {
    unsigned i = blockIdx.x * 256u + threadIdx.x;
    if (i >= KN) return;
    int4 c = coors[i];                        // (b, z, y, x)
    int slot = c.x * KSPATIAL + c.z * KNX + c.w;
    slot_arr[i] = slot;
    atomicMax(&winner[slot], (int)i);
}

// ---------------------------------------------------------------------------
// Kernel 4: 16 threads per voxel. Coalesced b128 nontemporal feature load,
// 4 scattered dword stores at stride NY*NX (forced by [B,C,NY,NX] layout).
// ---------------------------------------------------------------------------
__global__ __launch_bounds__(256) void scatter_kernel(const v4f* __restrict__ vf,
                                                      const int* __restrict__ winner,
                                                      const int* __restrict__ slot_arr,
                                                      float* __restrict__ out) {
    unsigned t = blockIdx.x * 256u + threadIdx.x;   // < KN * 16
    unsigned i = t >> 4;                            // voxel
    unsigned g = t & 15u;                           // channel group (4 ch)
    int slot = slot_arr[i];
    if (winner[slot] != (int)i) return;             // lost the duplicate race

    v4f v = __builtin_nontemporal_load(vf + ((size_t)i * 16 + g));

    unsigned b  = (unsigned)slot / KSPATIAL;
    unsigned sp = (unsigned)slot - b * KSPATIAL;    // y*NX + x
    size_t base = ((size_t)(b * KC + g * 4)) * KSPATIAL + sp;
    out[base]                        = v.x;
    out[base + (size_t)KSPATIAL]     = v.y;
    out[base + 2 * (size_t)KSPATIAL] = v.z;
    out[base + 3 * (size_t)KSPATIAL] = v.w;
}

// ---------------------------------------------------------------------------
// d_in[0] = voxel_features (float, N*C)
// d_in[1] = coors          (int,   N*4)
// d_in[2] = batch_size     (int scalar, == 8, baked in)
// d_out   = float [B, C, NY, NX]
// d_ws    = winner[KSLOTS] ints  +  slot_arr[KN] ints  (~7.1 MB)
// ---------------------------------------------------------------------------
extern "C" void kernel_launch(void* const* d_in, const int* in_sizes, int n_in,
                              void* d_out, int out_size, void* d_ws, size_t ws_size,
                              hipStream_t stream) {
    const v4f*  vf    = (const v4f*)d_in[0];
    const int4* coors = (const int4*)d_in[1];
    float* out        = (float*)d_out;

    int* winner   = (int*)d_ws;
    int* slot_arr = winner + KSLOTS;

    zero_fill_async<<<4096, 256, 0, stream>>>(out, (unsigned)KOUTGROUPS);
    init_winner<<<((KSLOTS / 4) + 255) / 256, 256, 0, stream>>>((v4i*)winner);
    claim_kernel<<<KN / 256, 256, 0, stream>>>(coors, winner, slot_arr);
    scatter_kernel<<<(KN * 16) / 256, 256, 0, stream>>>(vf, winner, slot_arr, out);
}